// Model_7078106104192
// MI455X (gfx1250) — compile-verified
//
#include <hip/hip_runtime.h>
#include <math.h>

// =====================================================================
// e3nn-style equivariant GNN forward for MI455X (gfx1250, wave32).
//  - fp32 throughout (reference dtype); per-node linears run on
//    V_WMMA_F32_16X16X4_F32 (16 nodes x 16 outputs, K stepped by 4).
//  - Branch-free WMMA operand feeds (clamped addresses + value selects).
//  - B fragments reused across the 2l+1 components (C accumulators/wave).
//  - CG coefficients computed on-device (fp64) replicating the reference.
//  - Edge tensor products skip dead l3>=3 channels; scatter uses native
//    global_atomic_add_f32 via unsafeAtomicAdd.
// Workspace (< 34 MB, fits in L2):
//   cg[2048] | consts[16] | sh[E*16] | nodesA[N*192] | nodesB[N*192] | agg[N*1456]
// Input mapping assumption (JAX sorted nested-dict flatten):
//   d_in[0]=positions d_in[1]=senders d_in[2]=receivers d_in[3..36]=param leaves
// =====================================================================

typedef float v2f __attribute__((ext_vector_type(2)));
typedef float v8f __attribute__((ext_vector_type(8)));

// ----------------- CG table offsets (floats) -----------------
#define OFF_110 0
#define OFF_111 9
#define OFF_112 36
#define OFF_011 81
#define OFF_022 90
#define OFF_121 115
#define OFF_122 160
#define OFF_132 235
#define OFF_211 340
#define OFF_212 385
#define OFF_220 460
#define OFF_221 485
#define OFF_222 560
#define OFF_231 685
#define OFF_232 790
#define OFF_213 965

struct Tri { int l1, l2, l3, off; };
__device__ const Tri CG_TRIPLES[16] = {
  {1,1,0,OFF_110},{1,1,1,OFF_111},{1,1,2,OFF_112},{0,1,1,OFF_011},
  {0,2,2,OFF_022},{1,2,1,OFF_121},{1,2,2,OFF_122},{1,3,2,OFF_132},
  {2,1,1,OFF_211},{2,1,2,OFF_212},{2,2,0,OFF_220},{2,2,1,OFF_221},
  {2,2,2,OFF_222},{2,3,1,OFF_231},{2,3,2,OFF_232},{2,1,3,OFF_213},
};

// ----------------- on-device CG construction (fp64) -----------------
struct C64 { double re, im; };

__device__ __forceinline__ double dfac(int n) {
  double r = 1.0;
  for (int i = 2; i <= n; ++i) r *= (double)i;
  return r;
}

__device__ double su2_elem(int j1, int j2, int j3, int m1, int m2) {
  int m3 = m1 + m2;
  if (m3 < -j3 || m3 > j3) return 0.0;
  double pref = sqrt((double)(2*j3+1) * dfac(j3+j1-j2) * dfac(j3-j1+j2) *
                     dfac(j1+j2-j3) / dfac(j1+j2+j3+1));
  pref *= sqrt(dfac(j3+m3)*dfac(j3-m3)*dfac(j1-m1)*dfac(j1+m1)*dfac(j2-m2)*dfac(j2+m2));
  double s = 0.0;
  for (int v = 0; v <= j1+j2-j3; ++v) {
    int a = j1+j2-j3-v, b = j1-m1-v, c = j2+m2-v, d = j3-j2+m1+v, e = j3-j1-m2+v;
    if (b < 0 || c < 0 || d < 0 || e < 0) continue;
    double term = 1.0 / (dfac(v)*dfac(a)*dfac(b)*dfac(c)*dfac(d)*dfac(e));
    s += (v & 1) ? -term : term;
  }
  return pref * s;
}

__device__ void build_r2c(int l, C64 q[7][7]) {
  const int n = 2*l+1;
  for (int i = 0; i < n; ++i)
    for (int j = 0; j < n; ++j) q[i][j] = {0.0, 0.0};
  const double ir2 = 0.7071067811865476;
  for (int m = -l; m < 0; ++m) {
    q[l+m][l-m] = {ir2, 0.0};
    q[l+m][l+m] = {0.0, -ir2};
  }
  q[l][l] = {1.0, 0.0};
  for (int m = 1; m <= l; ++m) {
    double sg = (m & 1) ? -1.0 : 1.0;
    q[l+m][l+m] = {sg*ir2, 0.0};
    q[l+m][l-m] = {0.0, sg*ir2};
  }
  double pr, pi;                       // (-i)^l
  switch (l & 3) {
    case 0: pr = 1; pi = 0; break;
    case 1: pr = 0; pi = -1; break;
    case 2: pr = -1; pi = 0; break;
    default: pr = 0; pi = 1; break;
  }
  for (int i = 0; i < n; ++i)
    for (int j = 0; j < n; ++j) {
      C64 v = q[i][j];
      q[i][j] = {v.re*pr - v.im*pi, v.re*pi + v.im*pr};
    }
}

__global__ void init_cg_kernel(float* __restrict__ cgt, float* __restrict__ cst) {
  const int t = threadIdx.x;
  if (t < 16) {
    const Tri tr = CG_TRIPLES[t];
    const int l1 = tr.l1, l2 = tr.l2, l3 = tr.l3, off = tr.off;
    const int n1 = 2*l1+1, n2 = 2*l2+1, n3 = 2*l3+1;
    C64 q1[7][7], q2[7][7], q3[7][7];
    build_r2c(l1, q1); build_r2c(l2, q2); build_r2c(l3, q3);
    double Cr[7][7][7], Ci[7][7][7];
    double bestMag = -1.0, phr = 1.0, phi = 0.0;
    for (int i = 0; i < n1; ++i)
      for (int j = 0; j < n2; ++j)
        for (int k = 0; k < n3; ++k) {
          double ar = 0.0, ai = 0.0;
          for (int a = 0; a < n1; ++a)
            for (int b = 0; b < n2; ++b) {
              int c = a + b - l1 - l2 + l3;   // m3 = m1 + m2
              if (c < 0 || c >= n3) continue;
              double su = su2_elem(l1, l2, l3, a - l1, b - l2);
              if (su == 0.0) continue;
              double pr = q1[a][i].re*q2[b][j].re - q1[a][i].im*q2[b][j].im;
              double pi = q1[a][i].re*q2[b][j].im + q1[a][i].im*q2[b][j].re;
              double rr = pr*q3[c][k].re + pi*q3[c][k].im;   // * conj(q3)
              double ri = pi*q3[c][k].re - pr*q3[c][k].im;
              ar += su*rr; ai += su*ri;
            }
          Cr[i][j][k] = ar; Ci[i][j][k] = ai;
          double mag = ar*ar + ai*ai;
          if (mag > bestMag) { bestMag = mag; phr = ar; phi = ai; }  // first max, ravel order
        }
    double pn = sqrt(phr*phr + phi*phi);
    if (pn < 1e-300) { phr = 1.0; phi = 0.0; pn = 1.0; }
    phr /= pn; phi /= pn;
    double ss = 0.0;
    for (int i = 0; i < n1; ++i)
      for (int j = 0; j < n2; ++j)
        for (int k = 0; k < n3; ++k) {
          double v = Cr[i][j][k]*phr + Ci[i][j][k]*phi;   // Re(Cc*conj(ph)/|ph|)
          Cr[i][j][k] = v; ss += v*v;
        }
    double sc = sqrt((double)n3 / ss);                    // component normalization
    for (int i = 0; i < n1; ++i)
      for (int j = 0; j < n2; ++j)
        for (int k = 0; k < n3; ++k)
          cgt[off + (i*n2 + j)*n3 + k] = (float)(Cr[i][j][k]*sc);
  }
  __threadfence();
  __syncthreads();
  if (t == 0) {
    double v0 = 0.276, v1 = -0.46, v2 = 0.843;
    double nr = sqrt(v0*v0 + v1*v1 + v2*v2);
    v0 /= nr; v1 /= nr; v2 /= nr;
    double y1[3] = { sqrt(3.0)*v1, sqrt(3.0)*v2, sqrt(3.0)*v0 };
    double y2[5];
    for (int k = 0; k < 5; ++k) {
      double s = 0.0;
      for (int i = 0; i < 3; ++i)
        for (int j = 0; j < 3; ++j)
          s += (double)cgt[OFF_112 + (i*3 + j)*5 + k]*y1[i]*y1[j];
      y2[k] = s;
    }
    double n2s = 0.0; for (int k = 0; k < 5; ++k) n2s += y2[k]*y2[k];
    double c2 = sqrt(5.0)/sqrt(n2s);
    double y3[7];
    for (int k = 0; k < 7; ++k) {
      double s = 0.0;
      for (int i = 0; i < 5; ++i)
        for (int j = 0; j < 3; ++j)
          s += (double)cgt[OFF_213 + (i*3 + j)*7 + k]*(c2*y2[i])*y1[j];
      y3[k] = s;
    }
    double n3s = 0.0; for (int k = 0; k < 7; ++k) n3s += y3[k]*y3[k];
    double c3 = sqrt(7.0)/sqrt(n3s);
    cst[0] = (float)c2; cst[1] = (float)c3;
  }
}

// ----------------- spherical harmonics per edge -----------------
__global__ void sh_kernel(const float* __restrict__ pos,
                          const int* __restrict__ snd, const int* __restrict__ rcv,
                          const float* __restrict__ cgt, const float* __restrict__ cst,
                          float* __restrict__ sh, int E) {
  int e = blockIdx.x*blockDim.x + threadIdx.x;
  if (e >= E) return;
  int s = snd[e], r = rcv[e];
  float dx = pos[r*3+0]-pos[s*3+0], dy = pos[r*3+1]-pos[s*3+1], dz = pos[r*3+2]-pos[s*3+2];
  float inv = rsqrtf(dx*dx + dy*dy + dz*dz);
  float ux = dx*inv, uy = dy*inv, uz = dz*inv;
  const float s3 = 1.7320508075688772f;
  float y1[3] = { s3*uy, s3*uz, s3*ux };
  float c2 = cst[0], c3 = cst[1];
  float y2[5];
  for (int k = 0; k < 5; ++k) {
    float acc = 0.f;
    for (int i = 0; i < 3; ++i)
      for (int j = 0; j < 3; ++j)
        acc += cgt[OFF_112 + (i*3 + j)*5 + k]*y1[i]*y1[j];
    y2[k] = c2*acc;
  }
  float y3[7];
  for (int k = 0; k < 7; ++k) {
    float acc = 0.f;
    for (int i = 0; i < 5; ++i)
      for (int j = 0; j < 3; ++j)
        acc += cgt[OFF_213 + (i*3 + j)*7 + k]*y2[i]*y1[j];
    y3[k] = c3*acc;
  }
  float* o = sh + (size_t)e*16;
  o[0]=y1[0]; o[1]=y1[1]; o[2]=y1[2];
  o[3]=y2[0]; o[4]=y2[1]; o[5]=y2[2]; o[6]=y2[3]; o[7]=y2[4];
  for (int k = 0; k < 7; ++k) o[8+k]=y3[k];
  o[15] = 0.f;
}

// ----------------- edge tensor-product + scatter -----------------
// l2==0 marks a direct copy of mul*(2l1+1) elements. src==-1 -> implicit 1.0 input.
struct EC { int src, dst, mul, l1, l2, l3, cg; };

// agg layout L0 (stride 40, K padded to 4): (0,1)@0, (1,-1)@4, (2,1)@16
__device__ const EC ECT0[3] = {
  {-1,  0, 1, 0, 0, 0, 0},
  {-1,  4, 1, 0, 1, 1, OFF_011},
  {-1, 16, 1, 0, 2, 2, OFF_022},
};
// node layout (192): s_e@0[32], s_o@32[32], v1e@64[8x3], v1o@88[8x3], v2e@112[8x5], v2o@152[8x5]
// agg layout L1 (1456): (0,1)@0[48] (0,-1)@48[48] (1,1)@96[80x3] (1,-1)@336[80x3]
//                       (2,1)@576[88x5] (2,-1)@1016[88x5]
__device__ const EC ECT1[36] = {
  {0,    0,  32, 0, 0, 0, 0},
  {88,  32,   8, 1, 1, 0, OFF_110},
  {112, 40,   8, 2, 2, 0, OFF_220},
  {32,  48,  32, 0, 0, 0, 0},
  {64,  80,   8, 1, 1, 0, OFF_110},
  {152, 88,   8, 2, 2, 0, OFF_220},
  {64,  96,   8, 1, 0, 0, 0},
  {32, 120,  32, 0, 1, 1, OFF_011},
  {88, 216,   8, 1, 1, 1, OFF_111},
  {64, 240,   8, 1, 2, 1, OFF_121},
  {152,264,   8, 2, 1, 1, OFF_211},
  {152,288,   8, 2, 3, 1, OFF_231},
  {112,312,   8, 2, 2, 1, OFF_221},
  {88, 336,   8, 1, 0, 0, 0},
  {0,  360,  32, 0, 1, 1, OFF_011},
  {88, 456,   8, 1, 2, 1, OFF_121},
  {64, 480,   8, 1, 1, 1, OFF_111},
  {152,504,   8, 2, 2, 1, OFF_221},
  {112,528,   8, 2, 1, 1, OFF_211},
  {112,552,   8, 2, 3, 1, OFF_231},
  {112,576,   8, 2, 0, 0, 0},
  {0,  616,  32, 0, 2, 2, OFF_022},
  {88, 776,   8, 1, 1, 2, OFF_112},
  {88, 816,   8, 1, 3, 2, OFF_132},
  {64, 856,   8, 1, 2, 2, OFF_122},
  {152,896,   8, 2, 1, 2, OFF_212},
  {152,936,   8, 2, 3, 2, OFF_232},
  {112,976,   8, 2, 2, 2, OFF_222},
  {152,1016,  8, 2, 0, 0, 0},
  {32, 1056, 32, 0, 2, 2, OFF_022},
  {88, 1216,  8, 1, 2, 2, OFF_122},
  {64, 1256,  8, 1, 1, 2, OFF_112},
  {64, 1296,  8, 1, 3, 2, OFF_132},
  {152,1336,  8, 2, 2, 2, OFF_222},
  {112,1376,  8, 2, 1, 2, OFF_212},
  {112,1416,  8, 2, 3, 2, OFF_232},
};
// agg layout L2 (96): (0,1)@0[48], (0,-1)@48[48]
__device__ const EC ECT2[6] = {
  {0,    0, 32, 0, 0, 0, 0},
  {88,  32,  8, 1, 1, 0, OFF_110},
  {112, 40,  8, 2, 2, 0, OFF_220},
  {32,  48, 32, 0, 0, 0, 0},
  {64,  80,  8, 1, 1, 0, OFF_110},
  {152, 88,  8, 2, 2, 0, OFF_220},
};

__global__ void edge_kernel(const float* __restrict__ nodes, int nstride,
                            const int* __restrict__ snd, const int* __restrict__ rcv,
                            const float* __restrict__ sh, const float* __restrict__ cgt,
                            float* __restrict__ agg, int astride, int layer, int E) {
  int e = blockIdx.x*blockDim.x + threadIdx.x;
  if (e >= E) return;
  const EC* tab; int nc;
  if (layer == 0)      { tab = ECT0; nc = 3; }
  else if (layer == 1) { tab = ECT1; nc = 36; }
  else                 { tab = ECT2; nc = 6; }
  int s = snd[e], r = rcv[e];
  float y[15];
  #pragma unroll
  for (int i = 0; i < 15; ++i) y[i] = sh[(size_t)e*16 + i];
  const float* xs = nodes ? (nodes + (size_t)s*nstride) : nullptr;
  float* ar = agg + (size_t)r*astride;
  for (int t = 0; t < nc; ++t) {
    EC c = tab[t];
    int d1 = 2*c.l1 + 1;
    if (c.l2 == 0) {                       // direct copy
      int n = c.mul*d1;
      for (int u = 0; u < n; ++u) {
        float v = xs ? xs[c.src + u] : 1.0f;
        unsafeAtomicAdd(&ar[c.dst + u], v);
      }
    } else {                               // CG contraction against y_{l2}
      int d2 = 2*c.l2 + 1, d3 = 2*c.l3 + 1;
      int yo = (c.l2 == 1) ? 0 : (c.l2 == 2) ? 3 : 8;
      const float* cg = cgt + c.cg;
      for (int mu = 0; mu < c.mul; ++mu)
        for (int k = 0; k < d3; ++k) {
          float acc = 0.f;
          for (int i = 0; i < d1; ++i) {
            float xv = xs ? xs[c.src + mu*d1 + i] : 1.0f;
            for (int j = 0; j < d2; ++j)
              acc += cg[(i*d2 + j)*d3 + k]*xv*y[yo + j];
          }
          unsafeAtomicAdd(&ar[c.dst + mu*d3 + k], acc);
        }
    }
  }
}

// ----------------- wave32 WMMA fp32 GEMM (branch-free feeds) -----------------
// For each 16-node x 16-output tile: D_m += A_m[16xKpad] * (B[KxO]*bscale),
// m = 0..C-1 share B.  A addresses are always valid by construction (rows are
// padded to Kpad); B uses clamped address + value select (no exec branches).
#define HP 200

template<int C>
__device__ __forceinline__ void chan_gemm(const float* Abase, int rStride,
                                          const float* __restrict__ W, int O,
                                          int K, int Kpad, float bscale,
                                          int lane, int ncol, v8f acc[C]) {
  const int row = lane & 15, hi = lane >> 4;
  const int bc  = (ncol < O) ? ncol : 0;           // clamped valid column
  const float bs = (ncol < O) ? bscale : 0.f;
  const float* ap0 = Abase + row*rStride;
  for (int k0 = 0; k0 < Kpad; k0 += 4) {
    const int ka = k0 + (hi << 1);
    const int i0 = (ka     < K) ? ka     : 0;      // clamped valid rows
    const int i1 = (ka + 1 < K) ? ka + 1 : 0;
    float b0 = W[i0*O + bc]*bs;
    float b1 = W[i1*O + bc]*bs;
    v2f b;
    b.x = (ka     < K) ? b0 : 0.f;
    b.y = (ka + 1 < K) ? b1 : 0.f;
    const float* ap = ap0 + ka*C;
    #pragma unroll
    for (int mm = 0; mm < C; ++mm) {
      v2f a;
      a.x = ap[mm];
      a.y = ap[C + mm];
      acc[mm] = __builtin_amdgcn_wmma_f32_16x16x4_f32(false, a, false, b, (short)0,
                                                      acc[mm], false, false);
    }
  }
}

// ----------------- normalized activations -----------------
__device__ __forceinline__ float gelu_n(float x) {
  const float k0 = 0.7978845608f;          // sqrt(2/pi)
  float g = 0.5f*x*(1.f + tanhf(k0*(x + 0.044715f*x*x*x)));
  return g*(1.0f/0.65196f);                // ~GELU_C
}
__device__ __forceinline__ float tanh_n(float x) { return tanhf(x)*(1.0f/0.62790f); }

// ----------------- per-node WMMA layer kernel (mid layers) -----------------
struct NodeArgs {
  const float* agg;  int astride;
  const float* prev; int pstride;
  float* out;        int ostride;
  const float* wpre[6];
  const float* wpost[6];
  const float* wsc[6];
  int AO[6]; int Kpre[6]; int Kpad[6]; int Ksc[6];
  float preScale[6]; float postScale[6]; float scScale[6];
};

template<int C, int O, int HO>
__device__ __forceinline__ void run_pre(const NodeArgs& A, int c, float* h,
                                        int n0, int lane) {
  const int col = lane & 15, hi = lane >> 4;
  const float* Ab = A.agg + n0*A.astride + A.AO[c];
  #pragma unroll
  for (int nt = 0; nt*16 < O; ++nt) {
    const int ncol = nt*16 + col;
    v8f acc[C];
    #pragma unroll
    for (int mm = 0; mm < C; ++mm) acc[mm] = v8f{0.f,0.f,0.f,0.f,0.f,0.f,0.f,0.f};
    if (A.Kpre[c] > 0)
      chan_gemm<C>(Ab, A.astride, A.wpre[c], O, A.Kpre[c], A.Kpad[c],
                   A.preScale[c], lane, ncol, acc);
    if (ncol < O) {
      #pragma unroll
      for (int mm = 0; mm < C; ++mm)
        #pragma unroll
        for (int r = 0; r < 8; ++r)
          h[(hi*8 + r)*HP + HO + ncol*C + mm] = acc[mm][r];
    }
  }
}

template<int C, int O, int HO>
__device__ __forceinline__ void run_post(const NodeArgs& A, int c, const float* h,
                                         int n0, int lane) {
  const int col = lane & 15, hi = lane >> 4;
  #pragma unroll
  for (int nt = 0; nt*16 < O; ++nt) {
    const int ncol = nt*16 + col;
    v8f acc[C];
    #pragma unroll
    for (int mm = 0; mm < C; ++mm) acc[mm] = v8f{0.f,0.f,0.f,0.f,0.f,0.f,0.f,0.f};
    chan_gemm<C>(h + HO, HP, A.wpost[c], O, O, O, A.postScale[c], lane, ncol, acc);
    if (A.Ksc[c] > 0)
      chan_gemm<C>(A.prev + n0*A.pstride + HO, A.pstride, A.wsc[c], O,
                   A.Ksc[c], A.Ksc[c], A.scScale[c], lane, ncol, acc);
    if (ncol < O) {
      float add = (A.Ksc[c] == -1) ? A.wsc[c][ncol]*A.scScale[c] : 0.f; // ones input
      #pragma unroll
      for (int mm = 0; mm < C; ++mm)
        #pragma unroll
        for (int r = 0; r < 8; ++r)
          A.out[(n0 + hi*8 + r)*A.ostride + HO + ncol*C + mm] = acc[mm][r] + add;
    }
  }
}

__global__ void __launch_bounds__(32)
node_mid_kernel(NodeArgs A) {
  __shared__ float h[16*HP];
  const int lane = threadIdx.x;
  const int n0   = blockIdx.x*16;

  // linear_pre -> h (LDS)
  run_pre<1, 32,   0>(A, 0, h, n0, lane);
  run_pre<1, 32,  32>(A, 1, h, n0, lane);
  run_pre<3,  8,  64>(A, 2, h, n0, lane);
  run_pre<3,  8,  88>(A, 3, h, n0, lane);
  run_pre<5,  8, 112>(A, 4, h, n0, lane);
  run_pre<5,  8, 152>(A, 5, h, n0, lane);
  __syncthreads();
  // scalar activations on l=0 blocks (cols 0..31 gelu, 32..63 tanh)
  for (int t = lane; t < 16*64; t += 32) {
    int nn = t >> 6, cc = t & 63;
    float v = h[nn*HP + cc];
    h[nn*HP + cc] = (cc < 32) ? gelu_n(v) : tanh_n(v);
  }
  __syncthreads();
  // linear_post + shortcut -> node features (global)
  run_post<1, 32,   0>(A, 0, h, n0, lane);
  run_post<1, 32,  32>(A, 1, h, n0, lane);
  run_post<3,  8,  64>(A, 2, h, n0, lane);
  run_post<3,  8,  88>(A, 3, h, n0, lane);
  run_post<5,  8, 112>(A, 4, h, n0, lane);
  run_post<5,  8, 152>(A, 5, h, n0, lane);
}

// ----------------- final layer (targets: 1x0o + 7x0e) -----------------
__global__ void node_final_kernel(const float* __restrict__ agg,   // [N][96]
                                  const float* __restrict__ prev,  // [N][192]
                                  const float* __restrict__ WpreO, const float* __restrict__ WpreE,
                                  const float* __restrict__ WpostO, const float* __restrict__ WpostE,
                                  const float* __restrict__ WscO,  const float* __restrict__ WscE,
                                  float* __restrict__ out, int N) {
  int n = blockIdx.x*blockDim.x + threadIdx.x;
  if (n >= N) return;
  const float* a = agg + (size_t)n*96;
  const float* x = prev + (size_t)n*192;
  const float se = 0.70710678f*0.14433757f;   // invAvg / sqrt(48)
  float he[7]; float ho;
  for (int o = 0; o < 7; ++o) {
    float s = 0.f;
    for (int i = 0; i < 48; ++i) s += a[i]*WpreE[i*7 + o];
    he[o] = s*se;
  }
  { float s = 0.f; for (int i = 0; i < 48; ++i) s += a[48 + i]*WpreO[i]; ho = s*se; }
  for (int o = 0; o < 7; ++o) he[o] = gelu_n(he[o]);
  ho = tanh_n(ho);
  float oe[7];
  const float sp7 = 0.37796447f;              // 1/sqrt(7)
  for (int o = 0; o < 7; ++o) {
    float s = 0.f;
    for (int i = 0; i < 7; ++i) s += he[i]*WpostE[i*7 + o];
    oe[o] = s*sp7;
  }
  float oo = ho*WpostO[0];
  const float s32 = 0.17677669f;              // 1/sqrt(32)
  for (int o = 0; o < 7; ++o) {
    float s = 0.f;
    for (int i = 0; i < 32; ++i) s += x[i]*WscE[i*7 + o];
    oe[o] += s*s32;
  }
  { float s = 0.f; for (int i = 0; i < 32; ++i) s += x[32 + i]*WscO[i]; oo += s*s32; }
  out[(size_t)n*8 + 0] = oo;
  for (int o = 0; o < 7; ++o) out[(size_t)n*8 + 1 + o] = oe[o];
}

// ===================== host-side orchestration =====================
extern "C" void kernel_launch(void* const* d_in, const int* in_sizes, int n_in,
                              void* d_out, int out_size, void* d_ws, size_t ws_size,
                              hipStream_t stream) {
  if (n_in < 37) return;  // unexpected flatten; cannot proceed safely
  const float* pos = (const float*)d_in[0];
  const int* snd = (const int*)d_in[1];
  const int* rcv = (const int*)d_in[2];
  const float* P[40];
  for (int i = 3; i < n_in && i < 40; ++i) P[i] = (const float*)d_in[i];

  const int N = in_sizes[0]/3;
  const int E = in_sizes[1];

  float* ws  = (float*)d_ws;
  float* cgt = ws;                              // 2048
  float* cst = ws + 2048;                       // 16
  float* sh  = ws + 4096;                       // E*16
  float* nA  = sh + (size_t)E*16;               // N*192
  float* nB  = nA + (size_t)N*192;              // N*192
  float* agg = nB + (size_t)N*192;              // N*1456 (max)

  const float invAvg = 0.70710678118654752f;    // 1/sqrt(AVG_NUM_NEIGHBORS)

  init_cg_kernel<<<1, 32, 0, stream>>>(cgt, cst);
  sh_kernel<<<(E + 255)/256, 256, 0, stream>>>(pos, snd, rcv, cgt, cst, sh, E);

  // ---------------- layer 0 ----------------
  hipMemsetAsync(agg, 0, (size_t)N*40*sizeof(float), stream);
  edge_kernel<<<(E + 255)/256, 256, 0, stream>>>((const float*)nullptr, 0, snd, rcv,
                                                 sh, cgt, agg, 40, 0, E);
  {
    NodeArgs A{};
    A.agg = agg; A.astride = 40; A.prev = nullptr; A.pstride = 0;
    A.out = nA;  A.ostride = 192;
    const int Kpre[6] = {1, 0, 0, 1, 1, 0};
    const int AO[6]   = {0, 0, 0, 4, 16, 0};
    const int Ksc[6]  = {-1, 0, 0, 0, 0, 0};
    const float* wpost[6] = {P[4], P[3], P[6], P[5], P[8], P[7]};
    const float* wpre[6]  = {P[9], nullptr, nullptr, P[10], P[11], nullptr};
    const float* wsc[6]   = {P[12], nullptr, nullptr, nullptr, nullptr, nullptr};
    const int Om[6] = {32, 32, 8, 8, 8, 8};
    for (int c = 0; c < 6; ++c) {
      A.Kpre[c] = Kpre[c]; A.Kpad[c] = 4; A.AO[c] = AO[c]; A.Ksc[c] = Ksc[c];
      A.wpre[c] = wpre[c]; A.wpost[c] = wpost[c]; A.wsc[c] = wsc[c];
      A.preScale[c]  = (Kpre[c] > 0) ? invAvg/sqrtf((float)Kpre[c]) : 0.f;
      A.postScale[c] = 1.0f/sqrtf((float)Om[c]);
      A.scScale[c]   = (c == 0) ? 1.0f : 0.f;   // shortcut in_mul = 1
    }
    node_mid_kernel<<<N/16, 32, 0, stream>>>(A);
  }

  // ---------------- layer 1 ----------------
  hipMemsetAsync(agg, 0, (size_t)N*1456*sizeof(float), stream);
  edge_kernel<<<(E + 255)/256, 256, 0, stream>>>(nA, 192, snd, rcv, sh, cgt,
                                                 agg, 1456, 1, E);
  {
    NodeArgs A{};
    A.agg = agg; A.astride = 1456; A.prev = nA; A.pstride = 192;
    A.out = nB;  A.ostride = 192;
    const int Kpre[6] = {48, 48, 80, 80, 88, 88};
    const int AO[6]   = {0, 48, 96, 336, 576, 1016};
    const int Ksc[6]  = {32, 32, 8, 8, 8, 8};
    const float* wpost[6] = {P[14], P[13], P[16], P[15], P[18], P[17]};
    const float* wpre[6]  = {P[20], P[19], P[22], P[21], P[24], P[23]};
    const float* wsc[6]   = {P[26], P[25], P[28], P[27], P[30], P[29]};
    const int Om[6] = {32, 32, 8, 8, 8, 8};
    for (int c = 0; c < 6; ++c) {
      A.Kpre[c] = Kpre[c]; A.Kpad[c] = Kpre[c]; A.AO[c] = AO[c]; A.Ksc[c] = Ksc[c];
      A.wpre[c] = wpre[c]; A.wpost[c] = wpost[c]; A.wsc[c] = wsc[c];
      A.preScale[c]  = invAvg/sqrtf((float)Kpre[c]);
      A.postScale[c] = 1.0f/sqrtf((float)Om[c]);
      A.scScale[c]   = 1.0f/sqrtf((float)Ksc[c]);
    }
    node_mid_kernel<<<N/16, 32, 0, stream>>>(A);
  }

  // ---------------- layer 2 (output) ----------------
  hipMemsetAsync(agg, 0, (size_t)N*96*sizeof(float), stream);
  edge_kernel<<<(E + 255)/256, 256, 0, stream>>>(nB, 192, snd, rcv, sh, cgt,
                                                 agg, 96, 2, E);
  node_final_kernel<<<(N + 255)/256, 256, 0, stream>>>(
      agg, nB,
      P[33] /*pre (0,-1) 48x1*/, P[34] /*pre (0,1) 48x7*/,
      P[31] /*post (0,-1) 1x1*/, P[32] /*post (0,1) 7x7*/,
      P[35] /*sc (0,-1) 32x1*/,  P[36] /*sc (0,1) 32x7*/,
      (float*)d_out, N);
}